// MLPGateDGL_bad_18004502904919
// MI455X (gfx1250) — compile-verified
//
#include <hip/hip_runtime.h>

// ============================================================================
// DeepGate-style GNN on MI455X (gfx1250, wave32).
//
// Key optimization vs the dense JAX reference: only ~1/18 of nodes are
// "active" per (level, code) iteration and only edges with BOTH endpoints
// active contribute. We compact the active set and run the conv + GRU
// matmuls as 16-node WMMA tiles using V_WMMA_F32_16X16X4_F32 (exact f32).
//
// WMMA fragment layout assumptions (from CDNA5 ISA 7.12.2, f32 16x16x4):
//   A (16x4, v2f): lanes 0-15 -> M=lane, {K=k,K=k+1}; lanes 16-31 -> {K=k+2,K=k+3}
//   B (4x16, v2f): symmetric  : lanes 0-15 -> N=lane, rows {k,k+1}; hi lanes {k+2,k+3}
//   C/D (16x16, v8f): VGPR v -> row v (lanes 0-15) / row v+8 (lanes 16-31), col=lane%16
// ============================================================================

typedef __attribute__((ext_vector_type(2))) float v2f;
typedef __attribute__((ext_vector_type(8))) float v8f;

#define HDIM 128          // hidden size (reference H)
#define SXPAD 4           // LDS row padding (stride % 64 == 4 -> conflict free)

__device__ __forceinline__ float sigmoidf_(float v) {
    return 1.0f / (1.0f + __expf(-v));
}

__device__ __forceinline__ v8f wmma_f32x4(v2f a, v2f b, v8f c) {
    return __builtin_amdgcn_wmma_f32_16x16x4_f32(false, a, false, b, (short)0, c,
                                                 false, false);
}

// ---------------------------------------------------------------------------
// hs = hf = 1.0  (reference initializes both states to ones)
// ---------------------------------------------------------------------------
__global__ void init_state_kernel(float* __restrict__ out, long long n) {
    long long i = (long long)blockIdx.x * blockDim.x + threadIdx.x;
    if (i < n) out[i] = 1.0f;
}

__global__ void reset_count_kernel(int* cnt) {
    if (threadIdx.x == 0) *cnt = 0;
}

// ---------------------------------------------------------------------------
// Per-node: if active, zero degree + agg rows and append to active list.
// List order is atomic-nondeterministic but all downstream math is indexed by
// node id, so output values do not depend on the ordering.
// ---------------------------------------------------------------------------
__global__ void prep_kernel(const int* __restrict__ lvl, const int* __restrict__ gate,
                            int level, int code,
                            float* __restrict__ outd, float* __restrict__ ind,
                            float* __restrict__ agg_s, float* __restrict__ agg_f,
                            int* __restrict__ list, int* __restrict__ cnt, int N) {
    int i = blockIdx.x * blockDim.x + threadIdx.x;
    if (i >= N) return;
    if (lvl[i] != level || gate[i] != code) return;
    outd[i] = 0.0f;
    ind[i]  = 0.0f;
    size_t rs = (size_t)i * HDIM;
    size_t rf = (size_t)i * (2 * HDIM);
    for (int c = 0; c < HDIM; ++c)     agg_s[rs + c] = 0.0f;
    for (int c = 0; c < 2 * HDIM; ++c) agg_f[rf + c] = 0.0f;
    int slot = atomicAdd(cnt, 1);
    list[slot] = i;
}

// ---------------------------------------------------------------------------
// Degrees over active edges (both endpoints active).
// ---------------------------------------------------------------------------
__global__ void degree_kernel(const int* __restrict__ esrc, const int* __restrict__ edst,
                              const int* __restrict__ lvl, const int* __restrict__ gate,
                              int level, int code,
                              float* __restrict__ outd, float* __restrict__ ind, int E) {
    int e = blockIdx.x * blockDim.x + threadIdx.x;
    if (e >= E) return;
    int s = esrc[e], d = edst[e];
    if (lvl[s] == level && gate[s] == code && lvl[d] == level && gate[d] == code) {
        atomicAdd(&outd[s], 1.0f);
        atomicAdd(&ind[d], 1.0f);
    }
}

// ---------------------------------------------------------------------------
// Message aggregation: one wave per edge, lanes stripe the feature dim.
//   agg_s[dst] += x[src] * cs[src]
//   agg_f[dst] += concat(x[src], hs_old[src]) * cs[src]
// Runs BEFORE the hs update of this iteration, so hs == hs_old here.
// ---------------------------------------------------------------------------
__global__ __launch_bounds__(256)
void aggregate_kernel(const int* __restrict__ esrc, const int* __restrict__ edst,
                      const int* __restrict__ lvl, const int* __restrict__ gate,
                      int level, int code,
                      const float* __restrict__ outd,
                      const float* __restrict__ x, const float* __restrict__ hs,
                      float* __restrict__ agg_s, float* __restrict__ agg_f, int E) {
    int e = blockIdx.x * 8 + (threadIdx.x >> 5);
    if (e >= E) return;
    int s = esrc[e], d = edst[e];
    if (!(lvl[s] == level && gate[s] == code && lvl[d] == level && gate[d] == code))
        return;
    float cs = rsqrtf(fmaxf(outd[s], 1.0f));
    int lane = threadIdx.x & 31;
    size_t sr = (size_t)s * HDIM;
    size_t ds = (size_t)d * HDIM;
    size_t df = (size_t)d * (2 * HDIM);
    for (int c = lane; c < HDIM; c += 32) {
        float xv = x[sr + c] * cs;
        float hv = hs[sr + c] * cs;
        atomicAdd(&agg_s[ds + c], xv);
        atomicAdd(&agg_f[df + c], xv);
        atomicAdd(&agg_f[df + HDIM + c], hv);
    }
}

// ---------------------------------------------------------------------------
// Fused graph-conv + GRU over one tile of 16 active nodes.
//   msg = relu( (agg*ci) @ cw + cb )            [16 x 128], cw is (DIN x 128)
//   gi  = msg @ wih^T ; gh = h @ whh^T          [16 x 384 each], fused in regs
//   h'  = (1-z)*tanh(in + r*hn) + z*h           written only for valid rows
// Block = 256 threads = 8 waves; wave w owns output-feature chunk n0 = 16*w.
// All WMMA-bearing control flow is wave-uniform (EXEC stays all-ones).
// ---------------------------------------------------------------------------
template <int DIN>
__global__ __launch_bounds__(256)
void gru_kernel(const int* __restrict__ list, const int* __restrict__ cnt,
                const float* __restrict__ agg, const float* __restrict__ ind,
                float* __restrict__ h_state,
                const float* __restrict__ cw, const float* __restrict__ cb,
                const float* __restrict__ wih, const float* __restrict__ whh,
                const float* __restrict__ bih, const float* __restrict__ bhh) {
    constexpr int SXS = DIN + SXPAD;        // padded stride: % 64 == 4
    constexpr int SHS = HDIM + SXPAD;

    __shared__ float sX[16 * SXS];          // scaled agg (conv input)
    __shared__ float sH[16 * SHS];          // hidden-state tile
    __shared__ float sM[16 * SHS];          // conv output (GRU input)
    __shared__ int   sNode[16];

    const int tid  = threadIdx.x;
    const int nAct = *cnt;
    const int base = blockIdx.x * 16;
    if (base >= nAct) return;               // block-uniform

    if (tid < 16) {
        int m = base + tid;
        sNode[tid] = (m < nAct) ? list[m] : -1;
    }
    __syncthreads();

    // Stage inputs: zero-pad invalid rows so WMMA math is harmless there.
    for (int idx = tid; idx < 16 * DIN; idx += 256) {
        int r = idx / DIN, c = idx % DIN;
        int node = sNode[r];
        float v = 0.0f;
        if (node >= 0) {
            float ci = rsqrtf(fmaxf(ind[node], 1.0f));
            v = agg[(size_t)node * DIN + c] * ci;
        }
        sX[r * SXS + c] = v;
    }
    for (int idx = tid; idx < 16 * HDIM; idx += 256) {
        int r = idx >> 7, c = idx & (HDIM - 1);
        int node = sNode[r];
        sH[r * SHS + c] = (node >= 0) ? h_state[(size_t)node * HDIM + c] : 0.0f;
    }
    __syncthreads();

    const int wave = tid >> 5;
    const int lLo  = tid & 15;
    const int lHi  = (tid >> 4) & 1;
    const int n0   = wave * 16;

    // ---- GEMM1: conv ---------------------------------------------------
    {
        v8f acc = {};
        for (int k = 0; k < DIN; k += 4) {
            v2f a, b;
            a[0] = sX[lLo * SXS + k + 2 * lHi];
            a[1] = sX[lLo * SXS + k + 2 * lHi + 1];
            b[0] = cw[(size_t)(k + 2 * lHi) * HDIM + n0 + lLo];
            b[1] = cw[(size_t)(k + 2 * lHi + 1) * HDIM + n0 + lLo];
            acc = wmma_f32x4(a, b, acc);
        }
        float bias = cb[n0 + lLo];
#pragma unroll
        for (int v = 0; v < 8; ++v)
            sM[(v + 8 * lHi) * SHS + n0 + lLo] = fmaxf(acc[v] + bias, 0.0f);
    }
    __syncthreads();

    // ---- GEMM2 (x6) + GRU elementwise, fully register-resident ----------
    v8f gi[3] = {{}, {}, {}};
    v8f gh[3] = {{}, {}, {}};
    for (int k = 0; k < HDIM; k += 4) {
        v2f am, ah;
        am[0] = sM[lLo * SHS + k + 2 * lHi];
        am[1] = sM[lLo * SHS + k + 2 * lHi + 1];
        ah[0] = sH[lLo * SHS + k + 2 * lHi];
        ah[1] = sH[lLo * SHS + k + 2 * lHi + 1];
#pragma unroll
        for (int g = 0; g < 3; ++g) {
            // gi[n] = sum_k msg[k] * wih[n, k]  (wih is (384 x 128) row-major)
            size_t rowi = (size_t)(g * HDIM + n0 + lLo) * HDIM + k + 2 * lHi;
            v2f bi, bh;
            bi[0] = wih[rowi];     bi[1] = wih[rowi + 1];
            bh[0] = whh[rowi];     bh[1] = whh[rowi + 1];
            gi[g] = wmma_f32x4(am, bi, gi[g]);
            gh[g] = wmma_f32x4(ah, bh, gh[g]);
        }
    }

    const float biR = bih[n0 + lLo],            bhR = bhh[n0 + lLo];
    const float biZ = bih[HDIM + n0 + lLo],     bhZ = bhh[HDIM + n0 + lLo];
    const float biN = bih[2 * HDIM + n0 + lLo], bhN = bhh[2 * HDIM + n0 + lLo];

#pragma unroll
    for (int v = 0; v < 8; ++v) {
        int row = v + 8 * lHi;
        int node = sNode[row];
        if (node < 0) continue;
        float r  = sigmoidf_(gi[0][v] + biR + gh[0][v] + bhR);
        float z  = sigmoidf_(gi[1][v] + biZ + gh[1][v] + bhZ);
        float nt = tanhf(gi[2][v] + biN + r * (gh[2][v] + bhN));
        float h  = sH[row * SHS + n0 + lLo];
        h_state[(size_t)node * HDIM + n0 + lLo] = (1.0f - z) * nt + z * h;
    }
}

// ---------------------------------------------------------------------------
// Readouts: wave-per-row dot products with shuffle reduction.
// ---------------------------------------------------------------------------
__device__ __forceinline__ float wave_sum(float v) {
    for (int o = 16; o > 0; o >>= 1) v += __shfl_xor(v, o, 32);
    return v;
}

__global__ __launch_bounds__(256)
void prob_kernel(const float* __restrict__ hf, const float* __restrict__ w,
                 const float* __restrict__ b, float* __restrict__ prob, int N) {
    int i = blockIdx.x * 8 + (threadIdx.x >> 5);
    if (i >= N) return;
    int lane = threadIdx.x & 31;
    float p = 0.0f;
    for (int c = lane; c < HDIM; c += 32) p += hf[(size_t)i * HDIM + c] * w[c];
    p = wave_sum(p);
    if (lane == 0) prob[i] = sigmoidf_(p + b[0]);
}

__global__ __launch_bounds__(256)
void rc_kernel(const float* __restrict__ hs, const int* __restrict__ rc,
               const float* __restrict__ w, const float* __restrict__ b,
               float* __restrict__ out, int P) {
    int p = blockIdx.x * 8 + (threadIdx.x >> 5);
    if (p >= P) return;
    int a = rc[p], c2 = rc[P + p];
    int lane = threadIdx.x & 31;
    float acc = 0.0f;
    for (int c = lane; c < HDIM; c += 32)
        acc += hs[(size_t)a * HDIM + c] * w[c] + hs[(size_t)c2 * HDIM + c] * w[HDIM + c];
    acc = wave_sum(acc);
    if (lane == 0) out[p] = sigmoidf_(acc + b[0]);
}

// ---------------------------------------------------------------------------
extern "C" void kernel_launch(void* const* d_in, const int* in_sizes, int n_in,
                              void* d_out, int out_size, void* d_ws, size_t ws_size,
                              hipStream_t stream) {
    const float* x    = (const float*)d_in[0];
    const int* lvl    = (const int*)d_in[1];
    const int* gatev  = (const int*)d_in[2];
    const int* esrc   = (const int*)d_in[3];
    const int* edst   = (const int*)d_in[4];
    const int* rcidx  = (const int*)d_in[5];
    const float* rp_w = (const float*)d_in[30];
    const float* rp_b = (const float*)d_in[31];
    const float* rr_w = (const float*)d_in[32];
    const float* rr_b = (const float*)d_in[33];

    const int N = in_sizes[1];
    const int E = in_sizes[3];
    const int P = in_sizes[5] / 2;

    // Output layout: hs (N*H) | hf (N*H) | prob (N) | is_rc (P)
    float* hs   = (float*)d_out;
    float* hf   = hs + (size_t)N * HDIM;
    float* prob = hf + (size_t)N * HDIM;
    float* isrc = prob + N;

    // Workspace carve-out
    char* wsb = (char*)d_ws;
    size_t off = 0;
    auto carve = [&](size_t bytes) -> char* {
        char* p = wsb + off;
        off = (off + bytes + 255) & ~(size_t)255;
        return p;
    };
    float* outd  = (float*)carve((size_t)N * 4);
    float* ind   = (float*)carve((size_t)N * 4);
    float* agg_s = (float*)carve((size_t)N * HDIM * 4);
    float* agg_f = (float*)carve((size_t)N * 2 * HDIM * 4);
    int*   list  = (int*)carve((size_t)N * 4);
    int*   cnt   = (int*)carve(256);
    (void)ws_size; (void)n_in; (void)out_size;

    long long stateElems = 2LL * N * HDIM;
    init_state_kernel<<<(int)((stateElems + 255) / 256), 256, 0, stream>>>(hs, stateElems);

    const int tileBlocks = (N + 15) / 16;   // worst case; blocks self-trim vs *cnt
    for (int level = 1; level < 6; ++level) {
        for (int code = 1; code <= 2; ++code) {
            int base = (code == 1) ? 6 : 18;   // and_* / or_* parameter blocks
            const float* cw_s  = (const float*)d_in[base + 0];
            const float* cb_s  = (const float*)d_in[base + 1];
            const float* wih_s = (const float*)d_in[base + 2];
            const float* whh_s = (const float*)d_in[base + 3];
            const float* bih_s = (const float*)d_in[base + 4];
            const float* bhh_s = (const float*)d_in[base + 5];
            const float* cw_f  = (const float*)d_in[base + 6];
            const float* cb_f  = (const float*)d_in[base + 7];
            const float* wih_f = (const float*)d_in[base + 8];
            const float* whh_f = (const float*)d_in[base + 9];
            const float* bih_f = (const float*)d_in[base + 10];
            const float* bhh_f = (const float*)d_in[base + 11];

            reset_count_kernel<<<1, 32, 0, stream>>>(cnt);
            prep_kernel<<<(N + 255) / 256, 256, 0, stream>>>(
                lvl, gatev, level, code, outd, ind, agg_s, agg_f, list, cnt, N);
            degree_kernel<<<(E + 255) / 256, 256, 0, stream>>>(
                esrc, edst, lvl, gatev, level, code, outd, ind, E);
            aggregate_kernel<<<(E + 7) / 8, 256, 0, stream>>>(
                esrc, edst, lvl, gatev, level, code, outd, x, hs, agg_s, agg_f, E);
            // hs update (reads agg_s); agg_f already captured hs_old above.
            gru_kernel<HDIM><<<tileBlocks, 256, 0, stream>>>(
                list, cnt, agg_s, ind, hs, cw_s, cb_s, wih_s, whh_s, bih_s, bhh_s);
            gru_kernel<2 * HDIM><<<tileBlocks, 256, 0, stream>>>(
                list, cnt, agg_f, ind, hf, cw_f, cb_f, wih_f, whh_f, bih_f, bhh_f);
        }
    }

    prob_kernel<<<(N + 7) / 8, 256, 0, stream>>>(hf, rp_w, rp_b, prob, N);
    rc_kernel<<<(P + 7) / 8, 256, 0, stream>>>(hs, rcidx, rr_w, rr_b, isrc, P);
}